// Diffusion_29901562315154
// MI455X (gfx1250) — compile-verified
//
#include <hip/hip_runtime.h>
#include <stdint.h>

// ---------------------------------------------------------------------------
// Diffusion on graph adjacency, MI455X (gfx1250, wave32).
//
//   K1: zero 4MB adjacency bitmask in d_ws          (B*N*N bits)
//   K2: scatter edges into bitmask with atomicOr    (idempotent -> deterministic)
//   K3: streaming kernel: Threefry2x32 Gumbel sampling + backward posterior.
//       402 MB of NT B128 stores (write-bandwidth bound: ~17 us @ 23.3 TB/s).
//       Bitmask tiles staged into LDS via CDNA5 async DMA
//       (global_load_async_to_lds_b128 + s_wait_asynccnt).
//
// Compute reductions vs naive:
//  - JAX counter-pairing: threefry(p, p+n/2) yields the uniforms of elements
//    e and e+HALF_E simultaneously -> 1 threefry per output element.
//  - Gumbel argmax over 2 categories == exponential race q1*ex0 > q0*ex1
//    (ex = -log u) -> 2 logs/element instead of 4.
//  - likelihood*prior/evidence depends only on (a,s) -> 4-entry per-batch
//    table in SGPRs; 16 divides per block instead of 8 per thread.
// ---------------------------------------------------------------------------

#define NB 8
#define NN 2048
#define NE 32768
#define TSTEPS 1000

static constexpr unsigned ELEM_TOTAL  = (unsigned)NB * NN * NN;  // 33,554,432
static constexpr unsigned HALF_E      = ELEM_TOTAL / 2u;         // 16,777,216
static constexpr unsigned GUMBEL_HALF = ELEM_TOTAL;              // 67,108,864/2 uniforms

typedef float    v4f __attribute__((ext_vector_type(4)));
typedef unsigned v4u __attribute__((ext_vector_type(4)));

struct U2 { unsigned x, y; };

__device__ __forceinline__ unsigned rotl32(unsigned v, int r) {
  return __builtin_rotateleft32(v, (unsigned)r);
}

// Exact JAX threefry2x32 with key = jax.random.key(42) -> (k0,k1) = (0,42).
__device__ __forceinline__ U2 threefry2x32_42(unsigned x0, unsigned x1) {
  const unsigned k0 = 0u, k1 = 42u;
  const unsigned k2 = k0 ^ k1 ^ 0x1BD11BDAu;
  x0 += k0; x1 += k1;
#define TF_R4(a,b,c,d)                              \
  x0 += x1; x1 = rotl32(x1,a); x1 ^= x0;            \
  x0 += x1; x1 = rotl32(x1,b); x1 ^= x0;            \
  x0 += x1; x1 = rotl32(x1,c); x1 ^= x0;            \
  x0 += x1; x1 = rotl32(x1,d); x1 ^= x0;
  TF_R4(13,15,26, 6)  x0 += k1; x1 += k2 + 1u;
  TF_R4(17,29,16,24)  x0 += k2; x1 += k0 + 2u;
  TF_R4(13,15,26, 6)  x0 += k0; x1 += k1 + 3u;
  TF_R4(17,29,16,24)  x0 += k1; x1 += k2 + 4u;
  TF_R4(13,15,26, 6)  x0 += k2; x1 += k0 + 5u;
#undef TF_R4
  return {x0, x1};
}

// jax.random.uniform(minval=tiny, maxval=1).
__device__ __forceinline__ float bits_to_u01(unsigned bits) {
  float f = __uint_as_float((bits >> 9) | 0x3f800000u) - 1.0f;
  return fmaxf(f, 1.17549435e-38f);
}

// ---------------------------------------------------------------------------
// K1: zero the bitmask (1,048,576 dwords) with B128 stores.
__global__ void __launch_bounds__(256) zero_bitmask(v4u* __restrict__ bm4) {
  unsigned i = blockIdx.x * 256u + threadIdx.x;   // grid sized exactly
  v4u z = {0u, 0u, 0u, 0u};
  bm4[i] = z;
}

// ---------------------------------------------------------------------------
// K2: scatter edges: bit = b*N*N + src*N + dst.
__global__ void __launch_bounds__(256) scatter_edges(const int* __restrict__ ei,
                                                     unsigned* __restrict__ bm) {
  unsigned idx = blockIdx.x * 256u + threadIdx.x;     // < NB*NE = 262,144
  unsigned b = idx >> 15;                             // / NE
  unsigned e = idx & (NE - 1u);                       // % NE
  unsigned base = b * 2u * NE;
  unsigned src = (unsigned)ei[base + e];
  unsigned dst = (unsigned)ei[base + NE + e];
  unsigned bit = (b << 22) + src * (unsigned)NN + dst;
  atomicOr(&bm[bit >> 5], 1u << (bit & 31u));
}

// ---------------------------------------------------------------------------
// K3: each block covers 1024 consecutive elements of the LOW half [0,HALF_E)
// plus the Threefry-paired 1024 elements at +HALF_E (batch b+4).
__global__ void __launch_bounds__(256)
diffusion_main(const float* __restrict__ Qt,     // [T,2,2] row-major
               const int*   __restrict__ t_arr,  // [B]
               const unsigned* __restrict__ bm,  // adjacency bitmask
               float* __restrict__ out)          // [ELEM] fwd ++ [2*ELEM] bwd
{
  __shared__ unsigned s_bm[64];                  // 32 low words + 32 high words
  const unsigned tid    = threadIdx.x;
  const unsigned chunkL = blockIdx.x * 1024u;
  const unsigned chunkH = chunkL + HALF_E;

  // --- CDNA5 async DMA: stage 2 x 128B bitmask tiles into LDS --------------
  if (tid < 16u) {
    unsigned w   = (tid < 8u) ? ((chunkL >> 5) + tid * 4u)
                              : ((chunkH >> 5) + (tid - 8u) * 4u);
    unsigned lds = (unsigned)(uintptr_t)(&s_bm[tid * 4u]);
    unsigned long long ga = (unsigned long long)(uintptr_t)(bm + w);
    asm volatile("global_load_async_to_lds_b128 %0, %1, off"
                 :: "v"(lds), "v"(ga) : "memory");
    asm volatile("s_wait_asynccnt 0x0" ::: "memory");
  }
  __syncthreads();

  // --- per-batch constants (uniform within block -> SGPRs) -----------------
  const unsigned bL = chunkL >> 22;              // / (N*N), in [0,4)
  const unsigned bH = bL + 4u;
  const int tL  = t_arr[bL];
  const int tH  = t_arr[bH];
  const int tLp = (tL == 0) ? (TSTEPS - 1) : (tL - 1);  // Qt[t-1] neg-index wrap
  const int tHp = (tH == 0) ? (TSTEPS - 1) : (tH - 1);
  const float nb0 = Qt[0],      fb0 = Qt[1];            // Qt[0]   row 0
  const float npL = Qt[4*tL],   pL  = Qt[4*tL + 1];     // Qt[t]   row 0 (low)
  const float npLp = Qt[4*tLp], pLp = Qt[4*tLp + 1];    // Qt[t-1] row 0 (low)
  const float npH = Qt[4*tH],   pH  = Qt[4*tH + 1];     // (high batch)
  const float npHp = Qt[4*tHp], pHp = Qt[4*tHp + 1];

  // Posterior tables qb{c}[a][s] = Qt[0][s][c] * Qt[t-1][a][c] / Qt[t][a][s].
  // Named scalars (not arrays) so they stay in SGPRs. ev(a,s): a==s ? np : p.
  const float qb0L_00 = (nb0*npLp)/npL, qb1L_00 = (fb0*pLp )/npL;
  const float qb0L_01 = (fb0*npLp)/pL , qb1L_01 = (nb0*pLp )/pL;
  const float qb0L_10 = (nb0*pLp )/pL , qb1L_10 = (fb0*npLp)/pL;
  const float qb0L_11 = (fb0*pLp )/npL, qb1L_11 = (nb0*npLp)/npL;
  const float qb0H_00 = (nb0*npHp)/npH, qb1H_00 = (fb0*pHp )/npH;
  const float qb0H_01 = (fb0*npHp)/pH , qb1H_01 = (nb0*pHp )/pH;
  const float qb0H_10 = (nb0*pHp )/pH , qb1H_10 = (fb0*npHp)/pH;
  const float qb0H_11 = (fb0*pHp )/npH, qb1H_11 = (nb0*npHp)/npH;

  const unsigned eL0   = chunkL + tid * 4u;
  const unsigned wbL   = s_bm[tid >> 3];
  const unsigned wbH   = s_bm[32u + (tid >> 3)];
  const unsigned shift = (tid & 7u) * 4u;

  v4f fwdL, fwdH;
  float bwL[8], bwH[8];

#pragma unroll
  for (int k = 0; k < 4; ++k) {
    const unsigned e = eL0 + (unsigned)k;        // low-half element index
    // bits[p<half] = threefry(p, p+half).x ; bits[p>=half] = threefry(p-half,p).y
    U2 r0 = threefry2x32_42(2u*e,      2u*e      + GUMBEL_HALF);
    U2 r1 = threefry2x32_42(2u*e + 1u, 2u*e + 1u + GUMBEL_HALF);
    float exL0 = -__logf(bits_to_u01(r0.x));
    float exH0 = -__logf(bits_to_u01(r0.y));
    float exL1 = -__logf(bits_to_u01(r1.x));
    float exH1 = -__logf(bits_to_u01(r1.y));

    { // low element (batch bL)
      bool a = ((wbL >> (shift + (unsigned)k)) & 1u) != 0u;
      float q0 = a ? pL : npL,  q1 = a ? npL : pL;       // Qt[t][a][:]
      bool  s  = (q1 * exL0) > (q0 * exL1);              // adj_noisy
      fwdL[k]    = s ? 1.0f : 0.0f;
      bwL[2*k]   = a ? (s ? qb0L_11 : qb0L_10) : (s ? qb0L_01 : qb0L_00);
      bwL[2*k+1] = a ? (s ? qb1L_11 : qb1L_10) : (s ? qb1L_01 : qb1L_00);
    }
    { // high element (batch bH = bL+4)
      bool a = ((wbH >> (shift + (unsigned)k)) & 1u) != 0u;
      float q0 = a ? pH : npH,  q1 = a ? npH : pH;
      bool  s  = (q1 * exH0) > (q0 * exH1);
      fwdH[k]    = s ? 1.0f : 0.0f;
      bwH[2*k]   = a ? (s ? qb0H_11 : qb0H_10) : (s ? qb0H_01 : qb0H_00);
      bwH[2*k+1] = a ? (s ? qb1H_11 : qb1H_10) : (s ? qb1H_01 : qb1H_00);
    }
  }

  // --- streaming NT stores: 402 MB, never re-read; keep it out of L2 -------
  float* out_fwd = out;
  float* out_bwd = out + ELEM_TOTAL;
  const unsigned eH0 = eL0 + HALF_E;
  __builtin_nontemporal_store(fwdL, (v4f*)(out_fwd + eL0));
  __builtin_nontemporal_store(fwdH, (v4f*)(out_fwd + eH0));
  __builtin_nontemporal_store(*(v4f*)&bwL[0], (v4f*)(out_bwd + 2u*eL0));
  __builtin_nontemporal_store(*(v4f*)&bwL[4], (v4f*)(out_bwd + 2u*eL0 + 4u));
  __builtin_nontemporal_store(*(v4f*)&bwH[0], (v4f*)(out_bwd + 2u*eH0));
  __builtin_nontemporal_store(*(v4f*)&bwH[4], (v4f*)(out_bwd + 2u*eH0 + 4u));
}

// ---------------------------------------------------------------------------
extern "C" void kernel_launch(void* const* d_in, const int* in_sizes, int n_in,
                              void* d_out, int out_size, void* d_ws, size_t ws_size,
                              hipStream_t stream) {
  (void)in_sizes; (void)n_in; (void)out_size; (void)ws_size;
  const float* Qt = (const float*)d_in[0];   // [T,2,2]
  const int*   ei = (const int*)d_in[1];     // [B,2,E]
  const int*   t  = (const int*)d_in[2];     // [B]
  unsigned*    bm = (unsigned*)d_ws;         // 4 MB adjacency bitmask
  float*      out = (float*)d_out;

  zero_bitmask<<<1024, 256, 0, stream>>>((v4u*)bm);     // 1,048,576 dwords
  scatter_edges<<<1024, 256, 0, stream>>>(ei, bm);      // 262,144 edges
  diffusion_main<<<16384, 256, 0, stream>>>(Qt, t, bm, out);
}